// EMKDLoss_73461120630840
// MI455X (gfx1250) — compile-verified
//
#include <hip/hip_runtime.h>
#include <math.h>

// ---------------------------------------------------------------------------
// EMKD loss for MI455X (gfx1250, wave32).
// Heavy work = two fp32 GEMMs (512x1024x32000 and 512x2048x32000) done with
// V_WMMA_F32_16X16X4_F32. GEMM is templated on K with N constexpr so all hot
// loads fold into base-pointer + 24-bit immediate offsets (no per-load u64
// address math). Everything else (Hungarian matching, reductions,
// cosine/smooth-L1, contrastive, projection MSE) is bandwidth-trivial.
// ---------------------------------------------------------------------------

typedef float v2f __attribute__((ext_vector_type(2)));
typedef float v8f __attribute__((ext_vector_type(8)));

#define NB    16          // 2*B batches after qry/pos concat
#define NVV   32          // nv
#define NTT   64          // nt
#define DS_   1024
#define DT_   2048
#define VV    32000
#define MROWS (NB * NVV)  // 512
#define TILE_K 64
#define LDS_STRIDE 66     // 64 + 2: conflict-free AND keeps 8-byte alignment

// ------------------------------ utility ------------------------------------

__global__ void zero_acc_kernel(float* acc) {
    if (threadIdx.x < 8) acc[threadIdx.x] = 0.0f;
}

// Gather sv (s_hid[:, :32]) -> A_s[512x1024] and tv (t_hid[:, 32:64]) -> A_t[512x2048]
__global__ void gather_kernel(const float* __restrict__ shq, const float* __restrict__ shp,
                              const float* __restrict__ thq, const float* __restrict__ thp,
                              float* __restrict__ A_s, float* __restrict__ A_t) {
    const int total = MROWS * DS_ + MROWS * DT_;
    for (int id = blockIdx.x * blockDim.x + threadIdx.x; id < total;
         id += gridDim.x * blockDim.x) {
        if (id < MROWS * DS_) {
            int row = id / DS_, k = id % DS_;
            int b = row / NVV, i = row % NVV;
            const float* src = (b < 8) ? (shq + (size_t)b * 96 * DS_)
                                       : (shp + (size_t)(b - 8) * 96 * DS_);
            A_s[id] = src[(size_t)i * DS_ + k];
        } else {
            int id2 = id - MROWS * DS_;
            int row = id2 / DT_, k = id2 % DT_;
            int b = row / NVV, i = row % NVV;
            const float* src = (b < 8) ? (thq + (size_t)b * 128 * DT_)
                                       : (thp + (size_t)(b - 8) * 128 * DT_);
            A_t[id2] = src[(size_t)(32 + i) * DT_ + k];   // tv = rows 32..63
        }
    }
}

// ------------------------------ WMMA GEMM ----------------------------------
// C[M x N] = A[M x K] * B[K x N], fp32, M = 512, N = 32000 (constexpr).
// Block tile: 64 (M) x 128 (N). 8 waves; wave w owns N strip w*16, with four
// 16x16 accumulators stacked in M (reuses each B fragment 4x).
// A-frag layout (ISA 7.12.2, 32-bit A 16x4): lane<16 holds K=k,k+1 for M=lane;
// lane>=16 holds K=k+2,k+3. B 4x16 mirrored; C/D: vgpr r = row half*8+r, col=lane&15.
template <int K>
__global__ __launch_bounds__(256) void wmma_gemm_f32(const float* __restrict__ A,
                                                     const float* __restrict__ Bm,
                                                     float* __restrict__ C) {
    constexpr int N = VV;
    __shared__ alignas(16) float Alds[64 * LDS_STRIDE];
    const int wave = threadIdx.x >> 5;
    const int lane = threadIdx.x & 31;
    const int half = lane >> 4;
    const int l16  = lane & 15;
    const int m0 = blockIdx.y * 64;
    const int nc = blockIdx.x * 128 + wave * 16 + l16;   // this lane's N column

    v8f acc[4];
#pragma unroll
    for (int ms = 0; ms < 4; ++ms) acc[ms] = (v8f){};

    // A-stage addressing: thread t covers (r0 + 4i, c0), i = 0..15
    const int r0 = threadIdx.x >> 6;
    const int c0 = threadIdx.x & 63;
    const float* ap = A + (size_t)(m0 + r0) * K + c0;    // advance by TILE_K per chunk
    float*       lp = &Alds[r0 * LDS_STRIDE + c0];

    // B addressing: one 64-bit base per lane; all loads in a chunk are
    // base + compile-time offset (< 8 MB, fits signed 24-bit IOFFSET)
    const float* bp = Bm + (size_t)(2 * half) * N + nc;

    for (int kk = 0; kk < K; kk += TILE_K) {
        // stage A tile 64 x TILE_K into LDS (4096 floats, 16 per thread)
#pragma unroll
        for (int i = 0; i < 16; ++i)
            lp[i * 4 * LDS_STRIDE] = ap[(size_t)i * 4 * K];
        __syncthreads();

        if (kk + TILE_K < K)   // stream-prefetch next B rows (global_prefetch_b8)
            __builtin_prefetch(bp + (size_t)TILE_K * N, 0, 0);

#pragma unroll
        for (int k4 = 0; k4 < TILE_K; k4 += 4) {
            const int kb = k4 + 2 * half;                 // even -> 8B-aligned LDS
            v2f b;
            b.x = bp[(size_t)k4 * N];
            b.y = bp[(size_t)k4 * N + N];
#pragma unroll
            for (int ms = 0; ms < 4; ++ms) {
                const v2f a = *(const v2f*)&Alds[(ms * 16 + l16) * LDS_STRIDE + kb];
                acc[ms] = __builtin_amdgcn_wmma_f32_16x16x4_f32(
                    false, a, false, b, (short)0, acc[ms], false, false);
            }
        }
        __syncthreads();
        ap += TILE_K;
        bp += (size_t)TILE_K * N;
    }

    // store: row = m0 + ms*16 + half*8 + r, col = nc
    float* cp = C + (size_t)(m0 + half * 8) * N + nc;
#pragma unroll
    for (int ms = 0; ms < 4; ++ms)
#pragma unroll
        for (int r = 0; r < 8; ++r)
            cp[(size_t)(ms * 16 + r) * N] = acc[ms][r];
}

// ----------------------- cost matrix + Hungarian ---------------------------

// cost[b][ti][sj] = sum_v |t_logits[b][ti][v] - s_logits[b][sj][v]|
__global__ void cost_kernel(const float* __restrict__ S, const float* __restrict__ T,
                            float* __restrict__ cost) {
    const int b = blockIdx.x, ti = blockIdx.y, sj = threadIdx.x;   // 32 threads
    const float* trow = T + (size_t)(b * NVV + ti) * VV;
    const float* srow = S + (size_t)(b * NVV + sj) * VV;
    float sum = 0.0f;
    for (int v = 0; v < VV; ++v) sum += fabsf(trow[v] - srow[v]);
    cost[(b * NVV + ti) * NVV + sj] = sum;
}

// Serial JV Hungarian per batch, f64 like the reference. 1 thread / block.
__global__ void hungarian_kernel(const float* __restrict__ cost, int* __restrict__ idx) {
    const int b = blockIdx.x;
    const int n = NVV;
    const float* C = cost + b * n * n;
    double u[NVV + 1], v[NVV + 1], minv[NVV + 1];
    int p[NVV + 1], way[NVV + 1];
    bool used[NVV + 1];
    for (int i = 0; i <= n; ++i) { u[i] = 0.0; v[i] = 0.0; p[i] = 0; way[i] = 0; }
    for (int i = 1; i <= n; ++i) {
        p[0] = i;
        int j0 = 0;
        for (int j = 0; j <= n; ++j) { minv[j] = 1e18; used[j] = false; }
        do {
            used[j0] = true;
            const int i0 = p[j0];
            double delta = 1e18;
            int j1 = -1;
            for (int j = 1; j <= n; ++j) {
                if (used[j]) continue;
                const double cur = (double)C[(i0 - 1) * n + (j - 1)] - u[i0] - v[j];
                if (cur < minv[j]) { minv[j] = cur; way[j] = j0; }
                if (minv[j] < delta) { delta = minv[j]; j1 = j; }
            }
            for (int j = 0; j <= n; ++j) {
                if (used[j]) { u[p[j]] += delta; v[j] -= delta; }
                else         { minv[j] -= delta; }
            }
            j0 = j1;
        } while (p[j0] != 0);
        while (j0) { const int j1 = way[j0]; p[j0] = p[j1]; j0 = j1; }
    }
    for (int j = 1; j <= n; ++j)
        if (p[j] > 0) idx[b * n + (p[j] - 1)] = j - 1;
}

// ------------------------------ reductions ---------------------------------

__device__ __forceinline__ void block_reduce_add(float val, float* target) {
    __shared__ float red[256];
    red[threadIdx.x] = val;
    __syncthreads();
    for (int s = blockDim.x >> 1; s > 0; s >>= 1) {
        if ((int)threadIdx.x < s) red[threadIdx.x] += red[threadIdx.x + s];
        __syncthreads();
    }
    if (threadIdx.x == 0) atomicAdd(target, red[0]);
}

// sum over b,r,v of (S[b][idx[b][r]][v] - T[b][r][v])^2  -> acc[1]
__global__ void vsd_kernel(const float* __restrict__ S, const float* __restrict__ T,
                           const int* __restrict__ idx, float* acc) {
    const size_t i = (size_t)blockIdx.x * blockDim.x + threadIdx.x;
    float val = 0.0f;
    if (i < (size_t)NB * NVV * VV) {
        const int v = (int)(i % VV);
        const int r = (int)((i / VV) % NVV);
        const int b = (int)(i / ((size_t)VV * NVV));
        const int sr = idx[b * NVV + r];
        const float d = S[(size_t)(b * NVV + sr) * VV + v] - T[i];
        val = d * d;
    }
    block_reduce_add(val, &acc[1]);
}

// row norms for sv (from A_s), st, tv (from A_t), tt
__global__ void norms_kernel(const float* __restrict__ A_s, const float* __restrict__ A_t,
                             const float* __restrict__ shq, const float* __restrict__ shp,
                             const float* __restrict__ thq, const float* __restrict__ thp,
                             float* n_sv, float* n_st, float* n_tv, float* n_tt) {
    const int id = blockIdx.x * blockDim.x + threadIdx.x;
    if (id >= NB * (NVV + NTT) * 2) return;
    const float* p;
    int len;
    float* dst;
    int slot;
    if (id < NB * NVV) {                               // sv
        p = A_s + (size_t)id * DS_; len = DS_; dst = n_sv; slot = id;
    } else if (id < NB * NVV + NB * NTT) {             // st = s_hid rows 32..95
        slot = id - NB * NVV;
        const int b = slot / NTT, j = slot % NTT;
        p = ((b < 8) ? shq + (size_t)b * 96 * DS_ : shp + (size_t)(b - 8) * 96 * DS_)
            + (size_t)(NVV + j) * DS_;
        len = DS_; dst = n_st;
    } else if (id < NB * NVV + NB * NTT + NB * NVV) {  // tv
        slot = id - NB * NVV - NB * NTT;
        p = A_t + (size_t)slot * DT_; len = DT_; dst = n_tv;
    } else {                                           // tt = t_hid rows 64..127
        slot = id - 2 * NB * NVV - NB * NTT;
        const int b = slot / NTT, j = slot % NTT;
        p = ((b < 8) ? thq + (size_t)b * 128 * DT_ : thp + (size_t)(b - 8) * 128 * DT_)
            + (size_t)(NTT + j) * DT_;
        len = DT_; dst = n_tt;
    }
    float s = 0.0f;
    for (int k = 0; k < len; ++k) s += p[k] * p[k];
    dst[slot] = sqrtf(s);
}

// smooth-L1( cos(sv_m, st) - cos(tv, tt) ), summed raw -> acc[2]
__global__ void vlad_kernel(const float* __restrict__ A_s, const float* __restrict__ A_t,
                            const float* __restrict__ shq, const float* __restrict__ shp,
                            const float* __restrict__ thq, const float* __restrict__ thp,
                            const int* __restrict__ idx,
                            const float* n_sv, const float* n_st,
                            const float* n_tv, const float* n_tt, float* acc) {
    const int gid = blockIdx.x * blockDim.x + threadIdx.x;   // NB*NVV*NTT = 32768
    float val = 0.0f;
    if (gid < NB * NVV * NTT) {
        const int j = gid & (NTT - 1);
        const int i = (gid >> 6) & (NVV - 1);
        const int b = gid >> 11;
        const int im = idx[b * NVV + i];
        const float* svm = A_s + (size_t)(b * NVV + im) * DS_;
        const float* st  = ((b < 8) ? shq + (size_t)b * 96 * DS_
                                    : shp + (size_t)(b - 8) * 96 * DS_)
                           + (size_t)(NVV + j) * DS_;
        const float* tv  = A_t + (size_t)(b * NVV + i) * DT_;
        const float* tt  = ((b < 8) ? thq + (size_t)b * 128 * DT_
                                    : thp + (size_t)(b - 8) * 128 * DT_)
                           + (size_t)(NTT + j) * DT_;
        float ds = 0.0f, dt = 0.0f;
        for (int k = 0; k < DS_; ++k) ds += svm[k] * st[k];
        for (int k = 0; k < DT_; ++k) dt += tv[k] * tt[k];
        const float cs = ds / (fmaxf(n_sv[b * NVV + im], 1e-8f) * fmaxf(n_st[b * NTT + j], 1e-8f));
        const float ct = dt / (fmaxf(n_tv[b * NVV + i],  1e-8f) * fmaxf(n_tt[b * NTT + j], 1e-8f));
        const float d = cs - ct, ad = fabsf(d);
        val = (ad < 1.0f) ? 0.5f * d * d : ad - 0.5f;
    }
    block_reduce_add(val, &acc[2]);
}

// -(1/8) * sum_i log_softmax(s_qry @ s_pos^T / 0.02)[i,i]  -> acc[3]
__global__ void contrastive_kernel(const float* __restrict__ sq,
                                   const float* __restrict__ sp, float* acc) {
    __shared__ float sc[64];
    const int i = threadIdx.x >> 3, j = threadIdx.x & 7;
    float d = 0.0f;
    for (int k = 0; k < DS_; ++k) d += sq[i * DS_ + k] * sp[j * DS_ + k];
    sc[i * 8 + j] = d * 50.0f;   // / TEMP (0.02)
    __syncthreads();
    if (threadIdx.x < 8) {
        const int r = threadIdx.x;
        float mx = -1e30f;
        for (int c = 0; c < 8; ++c) mx = fmaxf(mx, sc[r * 8 + c]);
        float se = 0.0f;
        for (int c = 0; c < 8; ++c) se += __expf(sc[r * 8 + c] - mx);
        const float lse = mx + __logf(se);
        atomicAdd(&acc[3], -(sc[r * 8 + r] - lse) * 0.125f);
    }
}

// sum of squared proj errors over qry+pos -> acc[0]
__global__ void mse_kernel(const float* __restrict__ s_qry, const float* __restrict__ s_pos,
                           const float* __restrict__ t_qry, const float* __restrict__ t_pos,
                           const float* __restrict__ Wp, const float* __restrict__ bp,
                           float* acc) {
    const int gid = blockIdx.x * blockDim.x + threadIdx.x;   // 2*8*1024
    float val = 0.0f;
    if (gid < 2 * 8 * DS_) {
        const int d = gid & (DS_ - 1);
        const int b = (gid >> 10) & 7;
        const int which = gid >> 13;
        const float* t = (which ? t_pos : t_qry) + (size_t)b * DT_;
        const float* s = (which ? s_pos : s_qry) + (size_t)b * DS_;
        float e = bp[d];
        for (int k = 0; k < DT_; ++k) e += t[k] * Wp[(size_t)k * DS_ + d];
        const float diff = s[d] - e;
        val = diff * diff;
    }
    block_reduce_add(val, &acc[0]);
}

__global__ void finalize_kernel(const float* __restrict__ acc, float* __restrict__ out) {
    if (threadIdx.x == 0) {
        const float contrastive = acc[3];
        const float mse  = acc[0] / 16384.0f;                 // 0.5*(S1/8192 + S2/8192)
        const float vsd  = acc[1] / (float)(NVV * VV);        // sum_b mean_(32,32000)
        const float vlad = acc[2] / (float)(NVV * NTT);       // sum_b mean_(32,64)
        const float distill = (vsd + vlad) / 8.0f;            // / Bn
        out[0] = contrastive + mse + distill;
        out[1] = contrastive;
        out[2] = distill;
    }
}

// ------------------------------ launcher -----------------------------------

extern "C" void kernel_launch(void* const* d_in, const int* in_sizes, int n_in,
                              void* d_out, int out_size, void* d_ws, size_t ws_size,
                              hipStream_t stream) {
    const float* s_qry = (const float*)d_in[0];
    const float* s_pos = (const float*)d_in[1];
    const float* t_qry = (const float*)d_in[2];
    const float* t_pos = (const float*)d_in[3];
    const float* shq   = (const float*)d_in[4];
    const float* shp   = (const float*)d_in[5];
    const float* thq   = (const float*)d_in[6];
    const float* thp   = (const float*)d_in[7];
    const float* W_s   = (const float*)d_in[8];
    const float* W_t   = (const float*)d_in[9];
    const float* W_p   = (const float*)d_in[10];
    const float* b_p   = (const float*)d_in[11];
    // nv (d_in[12]) and nt (d_in[13]) are compile-time constants here (32, 64)

    // workspace layout (floats); requires ~137.5 MB
    float* w    = (float*)d_ws;
    float* acc  = w;                                    // 8 (+pad to 16)
    float* A_s  = w + 16;                               // 512*1024
    float* A_t  = A_s + (size_t)MROWS * DS_;            // 512*2048
    float* Slog = A_t + (size_t)MROWS * DT_;            // 512*32000
    float* Tlog = Slog + (size_t)MROWS * VV;            // 512*32000
    float* cost = Tlog + (size_t)MROWS * VV;            // 16*32*32
    int*   idx  = (int*)(cost + NB * NVV * NVV);        // 512 ints
    float* n_sv = (float*)(idx + NB * NVV);             // 512
    float* n_st = n_sv + NB * NVV;                      // 1024
    float* n_tv = n_st + NB * NTT;                      // 512
    float* n_tt = n_tv + NB * NVV;                      // 1024

    zero_acc_kernel<<<1, 32, 0, stream>>>(acc);
    gather_kernel<<<6144, 256, 0, stream>>>(shq, shp, thq, thp, A_s, A_t);

    dim3 gg(VV / 128, MROWS / 64);                      // (250, 8)
    wmma_gemm_f32<DS_><<<gg, 256, 0, stream>>>(A_s, W_s, Slog);
    wmma_gemm_f32<DT_><<<gg, 256, 0, stream>>>(A_t, W_t, Tlog);

    cost_kernel<<<dim3(NB, NVV), NVV, 0, stream>>>(Slog, Tlog, cost);
    hungarian_kernel<<<NB, 1, 0, stream>>>(cost, idx);

    vsd_kernel<<<(NB * NVV * VV) / 256, 256, 0, stream>>>(Slog, Tlog, idx, acc);
    norms_kernel<<<12, 256, 0, stream>>>(A_s, A_t, shq, shp, thq, thp,
                                         n_sv, n_st, n_tv, n_tt);
    vlad_kernel<<<(NB * NVV * NTT) / 256, 256, 0, stream>>>(
        A_s, A_t, shq, shp, thq, thp, idx, n_sv, n_st, n_tv, n_tt, acc);
    contrastive_kernel<<<1, 64, 0, stream>>>(s_qry, s_pos, acc);
    mse_kernel<<<64, 256, 0, stream>>>(s_qry, s_pos, t_qry, t_pos, W_p, b_p, acc);

    finalize_kernel<<<1, 1, 0, stream>>>(acc, (float*)d_out);
}